// linearEncoder_70489003262325
// MI455X (gfx1250) — compile-verified
//
#include <hip/hip_runtime.h>

typedef float v8f __attribute__((ext_vector_type(8)));
typedef float v2f __attribute__((ext_vector_type(2)));

#define BTOT    2048
#define HIN     128
#define WINW    128
#define KDIM    4096     // 64*64 pooled features
#define NECK    512
#define KEEP    256
#define SLOTS   16
#define MTILE   16       // batch rows per block
#define KCHUNK  16
#define THREADS 256

__global__ __launch_bounds__(THREADS)
void fused_pool_gemm_topk(const float* __restrict__ x,
                          const float* __restrict__ W,
                          float* __restrict__ out)
{
    // Union LDS: [W slab 512x16 | A slab 16x16] during GEMM (33KB),
    // reused as feat tile 16x512 (32KB) afterwards.
    __shared__ float smem[NECK * KCHUNK + MTILE * KCHUNK];
    float* Wlds    = smem;                  // [NECK][KCHUNK]
    float* Alds    = smem + NECK * KCHUNK;  // [MTILE][KCHUNK]
    float* featLds = smem;                  // [MTILE][NECK] (aliases, post-GEMM)
    __shared__ unsigned threshLds[MTILE];

    const int tid    = threadIdx.x;
    const int wave   = tid >> 5;
    const int lane   = tid & 31;
    const int laneLo = lane & 15;
    const int khalf  = (lane >> 4) << 1;    // 0 for lanes 0-15, 2 for lanes 16-31
    const int b0     = blockIdx.x * MTILE;

    v8f acc[4] = {};                        // 4 N-tiles of 16x16 f32 per wave

    // A-staging role: one pooled element per thread per chunk
    const int aRow = tid >> 4;              // 0..15 (batch row in tile)
    const int aK   = tid & 15;              // k within chunk
    const long xBase = (long)(b0 + aRow) * (HIN * WINW);

    for (int k0 = 0; k0 < KDIM; k0 += KCHUNK) {
        // ---- stage A: 2x2 mean pool computed on the fly ----
        {
            int k = k0 + aK;
            int i = k >> 6, j = k & 63;
            const float* p0 = x + xBase + (2 * i) * WINW + 2 * j;
            float2 u0 = *(const float2*)p0;
            float2 u1 = *(const float2*)(p0 + WINW);
            Alds[aRow * KCHUNK + aK] = 0.25f * (u0.x + u0.y + u1.x + u1.y);
        }
        // ---- stage B: W[0:512][k0:k0+16], coalesced float4 ----
        {
            int rBase = tid >> 2;           // 0..63
            int f4    = (tid & 3) << 2;     // 0,4,8,12
#pragma unroll
            for (int it = 0; it < 8; ++it) {
                int n = rBase + it * 64;
                float4 v = *(const float4*)(W + (long)n * KDIM + k0 + f4);
                *(float4*)(Wlds + n * KCHUNK + f4) = v;
            }
        }
        // prefetch next W slab into cache hierarchy (global_prefetch_b8)
        if (k0 + KCHUNK < KDIM)
            __builtin_prefetch(W + (long)(tid >> 2) * KDIM + k0 + KCHUNK, 0, 1);

        __syncthreads();

        // ---- WMMA: D = A(16x4) x B(4x16) + C, f32 ----
#pragma unroll
        for (int kk = 0; kk < KCHUNK; kk += 4) {
            v2f a;
            a.x = Alds[laneLo * KCHUNK + kk + khalf];
            a.y = Alds[laneLo * KCHUNK + kk + khalf + 1];
#pragma unroll
            for (int t = 0; t < 4; ++t) {
                int n = wave * 64 + t * 16 + laneLo;
                v2f b;
                b.x = Wlds[n * KCHUNK + kk + khalf];
                b.y = Wlds[n * KCHUNK + kk + khalf + 1];
                acc[t] = __builtin_amdgcn_wmma_f32_16x16x4_f32(
                    /*neg_a=*/false, a, /*neg_b=*/false, b,
                    /*c_mod=*/(short)0, acc[t],
                    /*reuse_a=*/false, /*reuse_b=*/false);
            }
        }
        __syncthreads();  // before next chunk overwrites Alds/Wlds
    }

    // ---- spill accumulators to feat tile (C/D layout: vgpr r -> row r(+8)) ----
    {
        int rowOff = (lane >> 4) << 3;      // 0 or 8
#pragma unroll
        for (int t = 0; t < 4; ++t) {
            int n = wave * 64 + t * 16 + laneLo;
#pragma unroll
            for (int r = 0; r < 8; ++r)
                featLds[(r + rowOff) * NECK + n] = acc[t][r];
        }
    }
    __syncthreads();

    // ---- exact 256th-largest-|v| per row: MSB-first radix select on abs bits ----
    for (int rr = 0; rr < 2; ++rr) {
        int row = wave + rr * 8;            // 8 waves x 2 rows = 16
        unsigned u[16];
#pragma unroll
        for (int i = 0; i < 16; ++i) {
            float f = featLds[row * NECK + lane + 32 * i];
            u[i] = __float_as_uint(f) & 0x7fffffffu;
        }
        unsigned t = 0;
        for (int bit = 30; bit >= 0; --bit) {
            unsigned cand = t | (1u << bit);
            int cnt = 0;
#pragma unroll
            for (int i = 0; i < 16; ++i) cnt += (u[i] >= cand) ? 1 : 0;
#pragma unroll
            for (int off = 16; off >= 1; off >>= 1)
                cnt += __shfl_xor(cnt, off, 32);
            if (cnt >= KEEP) t = cand;      // max t with count(>=t) >= K  == kth largest
        }
        if (lane == 0) threshLds[row] = t;
    }
    __syncthreads();

    // ---- mask in place: keep |v| >= thresh (matches absf >= thresh incl. ties) ----
    for (int idx = tid; idx < MTILE * NECK; idx += THREADS) {
        int row = idx >> 9;
        float f = featLds[idx];
        unsigned a = __float_as_uint(f) & 0x7fffffffu;
        featLds[idx] = (a >= threshLds[row]) ? f : 0.0f;
    }
    __syncthreads();

    // ---- broadcast each row into 16 code slots, float4 stores ----
    const long outBase = (long)b0 * SLOTS * NECK;
    for (int idx = tid; idx < MTILE * SLOTS * (NECK / 4); idx += THREADS) {
        int row = idx >> 11;                // / (SLOTS * NECK/4)
        int rem = idx & 2047;
        int s   = rem >> 7;                 // slot
        int q   = (rem & 127) << 2;         // float4 column
        float4 v = *(const float4*)(featLds + row * NECK + q);
        *(float4*)(out + outBase + (long)(row * SLOTS + s) * NECK + q) = v;
    }
}

extern "C" void kernel_launch(void* const* d_in, const int* in_sizes, int n_in,
                              void* d_out, int out_size, void* d_ws, size_t ws_size,
                              hipStream_t stream) {
    const float* x = (const float*)d_in[0];
    const float* W = (const float*)d_in[1];
    float* out = (float*)d_out;
    fused_pool_gemm_topk<<<dim3(BTOT / MTILE), dim3(THREADS), 0, stream>>>(x, W, out);
}